// SDCN_53944789238432
// MI455X (gfx1250) — compile-verified
//
#include <hip/hip_runtime.h>
#include <hip/hip_bf16.h>

// SDCN forward on MI455X (gfx1250): fp32 WMMA (16x16x4) for all matmuls,
// double-buffered LDS tiles fed by GLOBAL_LOAD_ASYNC_TO_LDS (when the
// toolchain exposes the builtin), fused attention-row softmax in LDS.

#ifndef __has_builtin
#define __has_builtin(x) 0
#endif

#define HAS_ASYNC_LDS __has_builtin(__builtin_amdgcn_global_load_async_to_lds_b128)

typedef float v8f __attribute__((ext_vector_type(8)));
typedef float v2f __attribute__((ext_vector_type(2)));

#define NROWS 4096
#define BM 128
#define BN 64
#define BK 16

#if HAS_ASYNC_LDS
#define AS1 __attribute__((address_space(1)))
#define AS3 __attribute__((address_space(3)))
typedef int v4i __attribute__((ext_vector_type(4)));
__device__ __forceinline__ void cp_async16(const float* g, float* l) {
    // 16B per lane, memory -> LDS, tracked by ASYNCcnt.
    __builtin_amdgcn_global_load_async_to_lds_b128(
        (AS1 v4i*)const_cast<float*>(g), (AS3 v4i*)l, 0, 0);
}
#endif

__device__ __forceinline__ void wait_async_all() {
#if HAS_ASYNC_LDS
#if __has_builtin(__builtin_amdgcn_s_wait_asynccnt)
    __builtin_amdgcn_s_wait_asynccnt(0);
#else
    asm volatile("s_wait_asynccnt 0x0" ::: "memory");
#endif
#endif
}

__device__ __forceinline__ v8f wmma_f32(v2f a, v2f b, v8f c) {
    // D = A(16x4,f32) * B(4x16,f32) + C(16x16,f32)
    return __builtin_amdgcn_wmma_f32_16x16x4_f32(
        /*neg_a=*/false, a, /*neg_b=*/false, b,
        /*c_mod=*/(short)0, c, /*reuse_a=*/false, /*reuse_b=*/false);
}

// Generic tiled GEMM: C[4096,Nn] = act(A[4096,Kdim] @ B[Kdim,Nn] + bias)
// act: 0=none, 1=relu, 2=elu. 256 threads = 8 waves; wave w computes rows
// [w*16, w*16+16) of the BM=128 tile across all BN=64 columns.
// Double-buffered LDS; next tile streamed in with async-to-LDS copies while
// the current tile feeds 16 WMMAs per K-chunk.
__global__ __launch_bounds__(256)
void gemm_wmma_kernel(const float* __restrict__ A, const float* __restrict__ B,
                      const float* __restrict__ bias, float* __restrict__ C,
                      int Kdim, int Nn, int act)
{
    __shared__ __align__(16) float sA[2][BM * BK];
    __shared__ __align__(16) float sB[2][BK * BN];

    const int tid  = threadIdx.x;
    const int lane = tid & 31;
    const int wave = tid >> 5;
    const int hi   = (lane >> 4) & 1;   // K-pair select within fragment
    const int lo   = lane & 15;
    const int mBase = blockIdx.y * BM;
    const int nBase = blockIdx.x * BN;

    // B-tile copy coords (garbage beyond Nn is harmless: column n of C only
    // depends on column n of B, and columns >= Nn are never stored).
    const int bRow = tid >> 4;
    const int bC4  = (tid & 15) << 2;
    const int bN   = nBase + bC4;

    auto load_tiles = [&](int buf, int kBase) {
        // A tile (BM x BK): 512 float4, 2 per thread.
#pragma unroll
        for (int it = 0; it < 2; ++it) {
            const int idx = tid + it * 256;
            const int row = idx >> 2;
            const int c4  = (idx & 3) << 2;
            const float* g = A + (size_t)(mBase + row) * Kdim + kBase + c4;
#if HAS_ASYNC_LDS
            cp_async16(g, &sA[buf][idx << 2]);
#else
            *(float4*)(&sA[buf][idx << 2]) = *(const float4*)g;
#endif
        }
        // B tile (BK x BN): 256 float4, 1 per thread.
        if (bN < Nn) {
            const float* g = B + (size_t)(kBase + bRow) * Nn + bN;
#if HAS_ASYNC_LDS
            cp_async16(g, &sB[buf][tid << 2]);
#else
            *(float4*)(&sB[buf][tid << 2]) = *(const float4*)g;
#endif
        }
    };

    v8f acc[4];
#pragma unroll
    for (int t = 0; t < 4; ++t) acc[t] = (v8f){0.f,0.f,0.f,0.f,0.f,0.f,0.f,0.f};

    int buf = 0;
    load_tiles(0, 0);
    wait_async_all();
    __syncthreads();

    for (int kBase = 0; kBase < Kdim; kBase += BK) {
        const int kNext = kBase + BK;
        if (kNext < Kdim) load_tiles(buf ^ 1, kNext);
#if !HAS_ASYNC_LDS
        if (kNext + BK < Kdim) {
            // Pull the chunk-after-next toward L2 while we compute.
            __builtin_prefetch(A + (size_t)(mBase + (tid >> 2)) * Kdim + kNext + BK, 0, 0);
        }
#endif

#pragma unroll
        for (int k4 = 0; k4 < BK; k4 += 4) {
            v2f a;
            const int m = wave * 16 + lo;
            a.x = sA[buf][m * BK + k4 + 2 * hi];
            a.y = sA[buf][m * BK + k4 + 2 * hi + 1];
#pragma unroll
            for (int nt = 0; nt < 4; ++nt) {
                v2f b;
                const int n = nt * 16 + lo;
                b.x = sB[buf][(k4 + 2 * hi) * BN + n];
                b.y = sB[buf][(k4 + 2 * hi + 1) * BN + n];
                acc[nt] = wmma_f32(a, b, acc[nt]);
            }
        }

        wait_async_all();   // our async copies into buf^1 have landed
        __syncthreads();    // everyone done reading buf / writing buf^1
        buf ^= 1;
    }

    // Epilogue: C layout VGPR vi -> (M=vi | vi+8 by lane-hi, N=lane-lo).
#pragma unroll
    for (int nt = 0; nt < 4; ++nt) {
        const int col = nBase + nt * 16 + lo;
        if (col >= Nn) continue;
        const float bv = bias ? bias[col] : 0.f;
#pragma unroll
        for (int vi = 0; vi < 8; ++vi) {
            const int row = mBase + wave * 16 + vi + 8 * hi;
            float v = acc[nt][vi] + bv;
            if (act == 1)      v = fmaxf(v, 0.f);
            else if (act == 2) v = (v > 0.f) ? v : (__expf(v) - 1.f);
            C[(size_t)row * Nn + col] = v;
        }
    }
}

// s[i] = h[i,:] @ a_self ; t[i] = h[i,:] @ a_neigh. Wave per row (wave32).
__global__ __launch_bounds__(256)
void gemv_st_kernel(const float* __restrict__ h, const float* __restrict__ a_s,
                    const float* __restrict__ a_n, float* __restrict__ s,
                    float* __restrict__ t, int F)
{
    const int lane = threadIdx.x & 31;
    const int wave = threadIdx.x >> 5;
    const int row  = blockIdx.x * 8 + wave;
    const float* hr = h + (size_t)row * F;
    float vs = 0.f, vt = 0.f;
    for (int f = lane; f < F; f += 32) {
        const float hv = hr[f];
        vs += hv * a_s[f];
        vt += hv * a_n[f];
    }
    for (int off = 16; off > 0; off >>= 1) {
        vs += __shfl_down(vs, off, 32);
        vt += __shfl_down(vt, off, 32);
    }
    if (lane == 0) { s[row] = vs; t[row] = vt; }
}

// One block per row i: e_j = leaky((s_i + t_j)*M_ij), masked by adj, then
// row-softmax entirely in LDS (16KB e-row; 320KB/WGP available).
__global__ __launch_bounds__(256)
void attn_softmax_kernel(const float* __restrict__ s, const float* __restrict__ t,
                         const float* __restrict__ Mm, const float* __restrict__ adj,
                         float* __restrict__ att)
{
    __shared__ float ebuf[NROWS];
    __shared__ float red[256];
    const int tid = threadIdx.x;
    const int i   = blockIdx.x;
    const float si = s[i];
    const float* Mrow = Mm  + (size_t)i * NROWS;
    const float* arow = adj + (size_t)i * NROWS;

    float lmax = -3.4e38f;
    for (int j = tid; j < NROWS; j += 256) {
        __builtin_prefetch(Mrow + j + 1024, 0, 0);   // speculative, safe at tail
        __builtin_prefetch(arow + j + 1024, 0, 0);
        float e = (si + t[j]) * Mrow[j];
        e = (e > 0.f) ? e : 0.2f * e;              // LeakyReLU(0.2)
        e = (arow[j] > 0.f) ? e : -9.0e15f;        // adjacency mask
        ebuf[j] = e;
        lmax = fmaxf(lmax, e);
    }
    red[tid] = lmax; __syncthreads();
    for (int stp = 128; stp > 0; stp >>= 1) {
        if (tid < stp) red[tid] = fmaxf(red[tid], red[tid + stp]);
        __syncthreads();
    }
    const float mx = red[0];
    __syncthreads();

    float lsum = 0.f;
    for (int j = tid; j < NROWS; j += 256) {
        const float ex = __expf(ebuf[j] - mx);
        ebuf[j] = ex;
        lsum += ex;
    }
    red[tid] = lsum; __syncthreads();
    for (int stp = 128; stp > 0; stp >>= 1) {
        if (tid < stp) red[tid] += red[tid + stp];
        __syncthreads();
    }
    const float inv = 1.f / red[0];
    float* orow = att + (size_t)i * NROWS;
    for (int j = tid; j < NROWS; j += 256) orow[j] = ebuf[j] * inv;
}

__global__ void mix_kernel(float* __restrict__ out, const float* __restrict__ a,
                           const float* __restrict__ b, float sigma, int n,
                           float* __restrict__ out2)
{
    const int idx = blockIdx.x * blockDim.x + threadIdx.x;
    if (idx >= n) return;
    const float v = (1.f - sigma) * a[idx] + sigma * b[idx];
    out[idx] = v;
    if (out2) out2[idx] = v;
}

__global__ void softmax16_kernel(const float* __restrict__ in, float* __restrict__ out,
                                 int n)
{
    const int i = blockIdx.x * blockDim.x + threadIdx.x;
    if (i >= n) return;
    const float* r = in + (size_t)i * 16;
    float mx = r[0];
#pragma unroll
    for (int k = 1; k < 16; ++k) mx = fmaxf(mx, r[k]);
    float ex[16], sum = 0.f;
#pragma unroll
    for (int k = 0; k < 16; ++k) { ex[k] = __expf(r[k] - mx); sum += ex[k]; }
    const float inv = 1.f / sum;
    float* o = out + (size_t)i * 16;
#pragma unroll
    for (int k = 0; k < 16; ++k) o[k] = ex[k] * inv;
}

// Student-t: V=1 -> q_ik = (1+d_ik)^-1, row-normalized.
__global__ void studentt_kernel(const float* __restrict__ z,
                                const float* __restrict__ cluster,
                                float* __restrict__ q, int n)
{
    const int i = blockIdx.x * blockDim.x + threadIdx.x;
    if (i >= n) return;
    const float* zr = z + (size_t)i * 64;
    float qv[16], sum = 0.f;
#pragma unroll
    for (int k = 0; k < 16; ++k) {
        const float* c = cluster + k * 64;
        float d = 0.f;
        for (int f = 0; f < 64; ++f) { const float df = zr[f] - c[f]; d += df * df; }
        const float val = 1.f / (1.f + d);
        qv[k] = val; sum += val;
    }
    const float inv = 1.f / sum;
    float* o = q + (size_t)i * 16;
#pragma unroll
    for (int k = 0; k < 16; ++k) o[k] = qv[k] * inv;
}

extern "C" void kernel_launch(void* const* d_in, const int* in_sizes, int n_in,
                              void* d_out, int out_size, void* d_ws, size_t ws_size,
                              hipStream_t stream)
{
    (void)in_sizes; (void)n_in; (void)out_size; (void)ws_size;

    const float* x      = (const float*)d_in[0];
    const float* y      = (const float*)d_in[1];
    const float* adj    = (const float*)d_in[2];
    const float* Mm     = (const float*)d_in[3];
    const float* enc1_w = (const float*)d_in[4];
    const float* enc1_b = (const float*)d_in[5];
    const float* enc2_w = (const float*)d_in[6];
    const float* enc2_b = (const float*)d_in[7];
    const float* zl_w   = (const float*)d_in[8];
    const float* zl_b   = (const float*)d_in[9];
    const float* dec1_w = (const float*)d_in[10];
    const float* dec1_b = (const float*)d_in[11];
    const float* dec2_w = (const float*)d_in[12];
    const float* dec2_b = (const float*)d_in[13];
    const float* xbar_w = (const float*)d_in[14];
    const float* xbar_b = (const float*)d_in[15];
    const float* W1  = (const float*)d_in[16];
    const float* as1 = (const float*)d_in[17];
    const float* an1 = (const float*)d_in[18];
    const float* W2  = (const float*)d_in[19];
    const float* as2 = (const float*)d_in[20];
    const float* an2 = (const float*)d_in[21];
    const float* W3  = (const float*)d_in[22];
    const float* as3 = (const float*)d_in[23];
    const float* an3 = (const float*)d_in[24];
    const float* W4  = (const float*)d_in[25];
    const float* as4 = (const float*)d_in[26];
    const float* an4 = (const float*)d_in[27];
    const float* cluster = (const float*)d_in[28];

    // Workspace layout (floats), ~111 MB total.
    float* ws     = (float*)d_ws;
    float* att    = ws;                                   // 4096*4096
    float* tra1   = att    + (size_t)NROWS * NROWS;       // 4096*512
    float* tra2   = tra1   + (size_t)NROWS * 512;         // 4096*512
    float* zz     = tra2   + (size_t)NROWS * 512;         // 4096*64
    float* hbuf   = zz     + (size_t)NROWS * 64;          // 4096*512
    float* mixbuf = hbuf   + (size_t)NROWS * 512;         // 4096*512
    float* hWbuf  = mixbuf + (size_t)NROWS * 512;         // 4096*512
    float* zfbuf  = hWbuf  + (size_t)NROWS * 512;         // 4096*64
    float* svec   = zfbuf  + (size_t)NROWS * 64;          // 4096
    float* tvec   = svec   + NROWS;                       // 4096

    float* out      = (float*)d_out;
    float* out_xbar = out;                                // 4096*1024
    float* out_q    = out_xbar + (size_t)NROWS * 1024;    // 4096*16
    float* out_pred = out_q    + (size_t)NROWS * 16;      // 4096*16
    float* out_zf   = out_pred + (size_t)NROWS * 16;      // 4096*64

    auto gemm = [&](const float* A, const float* B, const float* bias, float* C,
                    int K, int N, int act) {
        dim3 grid((N + BN - 1) / BN, NROWS / BM);
        gemm_wmma_kernel<<<grid, 256, 0, stream>>>(A, B, bias, C, K, N, act);
    };
    auto gat = [&](const float* hin, int Fin, const float* W, const float* a_s,
                   const float* a_n, int Fout, float* hW, float* hout) {
        gemm(hin, W, nullptr, hW, Fin, Fout, /*act=*/0);
        gemv_st_kernel<<<NROWS / 8, 256, 0, stream>>>(hW, a_s, a_n, svec, tvec, Fout);
        attn_softmax_kernel<<<NROWS, 256, 0, stream>>>(svec, tvec, Mm, adj, att);
        gemm(att, hW, nullptr, hout, NROWS, Fout, /*act=elu*/2);
    };

    // --- Autoencoder ---
    gemm(x,      enc1_w, enc1_b, tra1,     1024, 512,  1);  // tra1 = relu
    gemm(tra1,   enc2_w, enc2_b, tra2,     512,  512,  1);  // tra2 = relu
    gemm(tra2,   zl_w,   zl_b,   zz,       512,  64,   0);  // z
    gemm(zz,     dec1_w, dec1_b, hbuf,     64,   512,  1);  // d1
    gemm(hbuf,   dec2_w, dec2_b, mixbuf,   512,  512,  1);  // d2
    gemm(mixbuf, xbar_w, xbar_b, out_xbar, 512,  1024, 0);  // x_bar

    const int NE = NROWS * 512;
    const float SIG = 0.5f;

    // --- GAT stack with AE feature injection ---
    gat(y, 1024, W1, as1, an1, 512, hWbuf, hbuf);                       // h1
    mix_kernel<<<(NE + 255) / 256, 256, 0, stream>>>(mixbuf, hbuf, tra1, SIG, NE, nullptr);
    gat(mixbuf, 512, W2, as2, an2, 512, hWbuf, hbuf);                   // h2
    mix_kernel<<<(NE + 255) / 256, 256, 0, stream>>>(mixbuf, hbuf, tra2, SIG, NE, nullptr);
    gat(mixbuf, 512, W3, as3, an3, 64, hWbuf, hbuf);                    // h3 [4096,64]
    const int NZF = NROWS * 64;
    mix_kernel<<<(NZF + 255) / 256, 256, 0, stream>>>(zfbuf, hbuf, zz, SIG, NZF, out_zf);
    gat(zfbuf, 64, W4, as4, an4, 16, hWbuf, hbuf);                      // h4 [4096,16]
    softmax16_kernel<<<(NROWS + 255) / 256, 256, 0, stream>>>(hbuf, out_pred, NROWS);

    // --- Student-t cluster assignment ---
    studentt_kernel<<<(NROWS + 255) / 256, 256, 0, stream>>>(zz, cluster, out_q, NROWS);
}